// GATModel_11742440587767
// MI455X (gfx1250) — compile-verified
//
#include <hip/hip_runtime.h>
#include <math.h>

#define BB 512
#define SS 128
#define CIN 32
#define HID 64
#define NH 8
#define HH (NH*HID)   /* 512 */
#define NEG_SLOPE 0.2f
#define NEG_INF -1000000000.0f

typedef __attribute__((ext_vector_type(2))) float v2f;
typedef __attribute__((ext_vector_type(8))) float v8f;

__device__ __forceinline__ v8f wmma_f32_k4(v2f a, v2f b, v8f c) {
  // D = A(16x4) * B(4x16) + C(16x16), fp32, wave32
  return __builtin_amdgcn_wmma_f32_16x16x4_f32(
      /*neg_a=*/false, a, /*neg_b=*/false, b,
      /*c_mod=*/(short)0, c, /*reuse_a=*/false, /*reuse_b=*/false);
}

// ---------------------------------------------------------------------------
// Kernel 1: h1 = x @ W1   (no bias; bias is added after attention per ref)
//           e_src1[b,s,h] = dot(h1[b,s,h*64:], a_src1[h])
//           e_dst1[b,s,h] = dot(h1[b,s,h*64:], a_dst1[h])
// grid = B*(S/16) blocks, 256 threads (8 waves = 8 heads)
// ---------------------------------------------------------------------------
__global__ __launch_bounds__(256) void gemm1_kernel(
    const float* __restrict__ x, const float* __restrict__ W1,
    const float* __restrict__ a_src1, const float* __restrict__ a_dst1,
    float* __restrict__ h1, float* __restrict__ e_src1, float* __restrict__ e_dst1)
{
  __shared__ float xs[16 * CIN];                 // 2 KB tile of x
  const int b  = blockIdx.x >> 3;
  const int s0 = (blockIdx.x & 7) << 4;
  for (int i = threadIdx.x; i < 16 * CIN; i += 256)
    xs[i] = x[(b * SS + s0) * CIN + i];          // 16 contiguous rows of 32
  __syncthreads();

  const int head = threadIdx.x >> 5;             // wave -> head
  const int lane = threadIdx.x & 31;
  const int nl = lane & 15, half = lane >> 4;

  // A fragments: lane nl = row M; half selects K-half of each K=4 step
  v2f afrag[8];
  #pragma unroll
  for (int kst = 0; kst < 8; ++kst) {
    const int k = kst * 4 + 2 * half;
    afrag[kst].x = xs[nl * CIN + k];
    afrag[kst].y = xs[nl * CIN + k + 1];
  }

  v8f acc[4];
  #pragma unroll
  for (int t = 0; t < 4; ++t) {
    v8f c = {0.f,0.f,0.f,0.f,0.f,0.f,0.f,0.f};
    const int col = head * HID + t * 16 + nl;
    #pragma unroll
    for (int kst = 0; kst < 8; ++kst) {
      const int k = kst * 4 + 2 * half;
      v2f bf;
      bf.x = W1[k * HH + col];
      bf.y = W1[(k + 1) * HH + col];
      c = wmma_f32_k4(afrag[kst], bf, c);
    }
    acc[t] = c;
  }

  // store h1: D layout -> M = r + 8*half, N = nl
  #pragma unroll
  for (int t = 0; t < 4; ++t) {
    const int col = head * HID + t * 16 + nl;
    #pragma unroll
    for (int r = 0; r < 8; ++r)
      h1[(size_t)(b * SS + s0 + r + 8 * half) * HH + col] = acc[t][r];
  }

  // fused attention-coefficient dots, reduced across the 16 lanes of each half
  float es[8], ed[8];
  #pragma unroll
  for (int r = 0; r < 8; ++r) { es[r] = 0.f; ed[r] = 0.f; }
  #pragma unroll
  for (int t = 0; t < 4; ++t) {
    const float as = a_src1[head * HID + t * 16 + nl];
    const float ad = a_dst1[head * HID + t * 16 + nl];
    #pragma unroll
    for (int r = 0; r < 8; ++r) { es[r] += acc[t][r] * as; ed[r] += acc[t][r] * ad; }
  }
  #pragma unroll
  for (int off = 1; off < 16; off <<= 1) {
    #pragma unroll
    for (int r = 0; r < 8; ++r) {
      es[r] += __shfl_xor(es[r], off, 32);
      ed[r] += __shfl_xor(ed[r], off, 32);
    }
  }
  if (nl == 0) {
    #pragma unroll
    for (int r = 0; r < 8; ++r) {
      const int srow = s0 + r + 8 * half;
      e_src1[(b * SS + srow) * NH + head] = es[r];
      e_dst1[(b * SS + srow) * NH + head] = ed[r];
    }
  }
}

// ---------------------------------------------------------------------------
// Kernel 2: layer-1 attention.  g1 = relu(softmax(leaky(e_s+e_d), mask) @ h1 + b1)
// grid = B*H*2 blocks, 128 threads (4 waves = 4 t-tiles of 16)
// LDS: 32 KB attn scratch + 32 KB staged h1 head panel = 64 KB
// ---------------------------------------------------------------------------
__global__ __launch_bounds__(128) void attn1_kernel(
    const float* __restrict__ h1, const float* __restrict__ e_src1,
    const float* __restrict__ e_dst1, const int* __restrict__ adj,
    const float* __restrict__ b1, float* __restrict__ g1)
{
  __shared__ float attnS[4 * 16 * SS];           // 32 KB: one 16x128 tile per wave
  __shared__ float hh[SS * HID];                 // 32 KB: h1[b, :, h*64:(h+1)*64]
  const int blk = blockIdx.x;
  const int tg = blk & 1;                        // which half of the 8 t-tiles
  const int h  = (blk >> 1) & 7;
  const int b  = blk >> 4;

  const int wave = threadIdx.x >> 5;
  const int lane = threadIdx.x & 31;
  const int nl = lane & 15, half = lane >> 4;
  const int t0 = (tg * 4 + wave) * 16;
  float* attn = attnS + wave * 16 * SS;

  // cooperative stage: head panel (coalesced float4) + e_src1 column.
  // e_src1 column is staged in attnS[0:128] and consumed before attn is written.
  const float* hsrc = h1 + (size_t)(b * SS) * HH + h * HID;
  for (int i = threadIdx.x; i < SS * HID / 4; i += 128) {
    const int s = i >> 4, q = i & 15;
    ((float4*)hh)[i] = *(const float4*)(hsrc + (size_t)s * HH + q * 4);
  }
  float* esL = attnS;                            // union: es staging area
  if (threadIdx.x < SS)
    esL[threadIdx.x] = e_src1[(b * SS + threadIdx.x) * NH + h];
  __syncthreads();

  // --- softmax over sources: lane nl owns target row t0+nl, half owns 64 sources
  const int t = t0 + nl;
  const float edt = e_dst1[(b * SS + t) * NH + h];
  const int* adjb = adj + (size_t)(b * 2 + 1) * SS * SS;   // adj[b,1,s,t]
  const int sbase = half * 64;

  float vals[64];
  float mx = -3.0e38f;
  #pragma unroll
  for (int i = 0; i < 64; ++i) {
    const int s = sbase + i;
    float l = esL[s] + edt;
    l = (l > 0.f) ? l : NEG_SLOPE * l;
    const bool m = (adjb[s * SS + t] != 0) || (s == t);
    l = m ? l : NEG_INF;
    vals[i] = l;
    mx = fmaxf(mx, l);
  }
  mx = fmaxf(mx, __shfl_xor(mx, 16, 32));
  float sum = 0.f;
  #pragma unroll
  for (int i = 0; i < 64; ++i) {
    const float e = __expf(vals[i] - mx);
    vals[i] = e;
    sum += e;
  }
  sum += __shfl_xor(sum, 16, 32);
  const float inv = 1.0f / sum;
  __syncthreads();                               // all esL reads done before overwrite
  #pragma unroll
  for (int i = 0; i < 64; ++i)
    attn[nl * SS + sbase + i] = vals[i] * inv;
  __syncthreads();

  // --- out(16x64) = attn(16x128) @ hh(128x64) via 32 K-steps of f32 WMMA
  v8f acc[4];
  #pragma unroll
  for (int tn = 0; tn < 4; ++tn) {
    v8f c = {0.f,0.f,0.f,0.f,0.f,0.f,0.f,0.f};
    const int col = tn * 16 + nl;
    for (int kst = 0; kst < 32; ++kst) {
      const int k = kst * 4 + 2 * half;
      v2f a, bf;
      a.x  = attn[nl * SS + k];
      a.y  = attn[nl * SS + k + 1];
      bf.x = hh[k * HID + col];
      bf.y = hh[(k + 1) * HID + col];
      c = wmma_f32_k4(a, bf, c);
    }
    acc[tn] = c;
  }

  #pragma unroll
  for (int tn = 0; tn < 4; ++tn) {
    const int col = tn * 16 + nl;
    const float bias = b1[h * HID + col];
    #pragma unroll
    for (int r = 0; r < 8; ++r) {
      const int tt = t0 + r + 8 * half;
      g1[(size_t)(b * SS + tt) * HH + h * HID + col] = fmaxf(acc[tn][r] + bias, 0.f);
    }
  }
}

// ---------------------------------------------------------------------------
// Kernel 3: h2 = g1 @ W2   ((B*S,512) @ (512,64))
// grid = B*S/16 blocks, 128 threads (4 waves = 4 N-tiles)
// LDS: 32 KB A-panel + 32 KB W2 K-chunk (staged 4x) = 64 KB
// ---------------------------------------------------------------------------
__global__ __launch_bounds__(128) void gemm2_kernel(
    const float* __restrict__ g1, const float* __restrict__ W2,
    float* __restrict__ h2)
{
  __shared__ float gs[16 * HH];                  // 32 KB A panel (full K)
  __shared__ float ws2[128 * HID];               // 32 KB W2 chunk (K=128)
  const size_t m0 = (size_t)blockIdx.x * 16;
  const float4* src = (const float4*)(g1 + m0 * HH);
  for (int i = threadIdx.x; i < 16 * HH / 4; i += 128)
    ((float4*)gs)[i] = src[i];

  const int wave = threadIdx.x >> 5;             // N-tile
  const int lane = threadIdx.x & 31;
  const int nl = lane & 15, half = lane >> 4;
  const int col = wave * 16 + nl;

  v8f c = {0.f,0.f,0.f,0.f,0.f,0.f,0.f,0.f};
  for (int c4 = 0; c4 < 4; ++c4) {
    __syncthreads();                             // prev chunk consumed (+ gs ready)
    const float4* wsrc = (const float4*)(W2 + c4 * 128 * HID);
    for (int i = threadIdx.x; i < 128 * HID / 4; i += 128)
      ((float4*)ws2)[i] = wsrc[i];
    __syncthreads();
    for (int kst = 0; kst < 32; ++kst) {
      const int kl = kst * 4 + 2 * half;
      v2f a, bf;
      a.x  = gs[nl * HH + c4 * 128 + kl];
      a.y  = gs[nl * HH + c4 * 128 + kl + 1];
      bf.x = ws2[kl * HID + col];
      bf.y = ws2[(kl + 1) * HID + col];
      c = wmma_f32_k4(a, bf, c);
    }
  }
  #pragma unroll
  for (int r = 0; r < 8; ++r)
    h2[(m0 + r + 8 * half) * HID + col] = c[r];
}

// ---------------------------------------------------------------------------
// Kernel 4: e_src2[b,s] = dot(h2[b,s], a_src2);  e_dst_last[b] at s = S-1 only
// ---------------------------------------------------------------------------
__global__ __launch_bounds__(256) void e2_kernel(
    const float* __restrict__ h2, const float* __restrict__ a_src2,
    const float* __restrict__ a_dst2, float* __restrict__ e_src2,
    float* __restrict__ e_dst_last)
{
  const int idx = blockIdx.x * 256 + threadIdx.x;
  if (idx >= BB * SS) return;
  const float* hr = h2 + (size_t)idx * HID;
  float es = 0.f, ed = 0.f;
  #pragma unroll
  for (int c = 0; c < HID; ++c) { const float v = hr[c]; es += v * a_src2[c]; ed += v * a_dst2[c]; }
  e_src2[idx] = es;
  if ((idx & (SS - 1)) == SS - 1) e_dst_last[idx / SS] = ed;
}

// ---------------------------------------------------------------------------
// Kernel 5: layer-2 attention only for t = S-1, then final linear + sigmoid.
// grid = B blocks, 128 threads
// ---------------------------------------------------------------------------
__global__ __launch_bounds__(128) void final_kernel(
    const float* __restrict__ h2, const float* __restrict__ e_src2,
    const float* __restrict__ e_dst_last, const int* __restrict__ adj,
    const float* __restrict__ b2, const float* __restrict__ Wf,
    const float* __restrict__ bfp, float* __restrict__ out)
{
  __shared__ float p[SS];
  __shared__ float red[4];
  const int b = blockIdx.x;
  const int tid = threadIdx.x;
  const int lane = tid & 31, wave = tid >> 5;

  const float edl = e_dst_last[b];
  const int* adjb = adj + (size_t)(b * 2 + 1) * SS * SS;

  // logits for source s = tid at target t = S-1
  const int s = tid;
  float l = e_src2[b * SS + s] + edl;
  l = (l > 0.f) ? l : NEG_SLOPE * l;
  const bool m = (adjb[s * SS + (SS - 1)] != 0) || (s == SS - 1);
  l = m ? l : NEG_INF;

  float mx = l;
  #pragma unroll
  for (int off = 16; off >= 1; off >>= 1) mx = fmaxf(mx, __shfl_xor(mx, off, 32));
  if (lane == 0) red[wave] = mx;
  __syncthreads();
  mx = fmaxf(fmaxf(red[0], red[1]), fmaxf(red[2], red[3]));

  const float e = __expf(l - mx);
  float sum = e;
  #pragma unroll
  for (int off = 16; off >= 1; off >>= 1) sum += __shfl_xor(sum, off, 32);
  __syncthreads();
  if (lane == 0) red[wave] = sum;
  __syncthreads();
  const float tot = red[0] + red[1] + red[2] + red[3];
  p[s] = e / tot;
  __syncthreads();

  // weighted sum + bias + relu + final dot
  float contrib = 0.f;
  if (tid < HID) {
    float acc = 0.f;
    for (int ss2 = 0; ss2 < SS; ++ss2)
      acc += p[ss2] * h2[(size_t)(b * SS + ss2) * HID + tid];
    const float v = fmaxf(acc + b2[tid], 0.f);
    contrib = v * Wf[tid];
  }
  #pragma unroll
  for (int off = 16; off >= 1; off >>= 1) contrib += __shfl_xor(contrib, off, 32);
  if (lane == 0) red[wave] = contrib;   // waves 2,3 contribute 0
  __syncthreads();
  if (tid == 0) {
    const float z = red[0] + red[1] + bfp[0];
    out[b] = 1.0f / (1.0f + __expf(-z));
  }
}

// ---------------------------------------------------------------------------
extern "C" void kernel_launch(void* const* d_in, const int* in_sizes, int n_in,
                              void* d_out, int out_size, void* d_ws, size_t ws_size,
                              hipStream_t stream)
{
  const float* x      = (const float*)d_in[0];
  const int*   adj    = (const int*)  d_in[1];
  const float* W1     = (const float*)d_in[2];
  const float* a_src1 = (const float*)d_in[3];
  const float* a_dst1 = (const float*)d_in[4];
  const float* b1     = (const float*)d_in[5];
  const float* W2     = (const float*)d_in[6];
  const float* a_src2 = (const float*)d_in[7];
  const float* a_dst2 = (const float*)d_in[8];
  const float* b2     = (const float*)d_in[9];
  const float* Wf     = (const float*)d_in[10];
  const float* bfp    = (const float*)d_in[11];
  float* out = (float*)d_out;

  float* ws = (float*)d_ws;
  size_t off = 0;
  float* h1 = ws + off;         off += (size_t)BB * SS * HH;    // 128 MB
  float* g1 = ws + off;         off += (size_t)BB * SS * HH;    // 128 MB
  float* h2 = ws + off;         off += (size_t)BB * SS * HID;   // 16 MB
  float* e_src1 = ws + off;     off += (size_t)BB * SS * NH;
  float* e_dst1 = ws + off;     off += (size_t)BB * SS * NH;
  float* e_src2 = ws + off;     off += (size_t)BB * SS;
  float* e_dst_last = ws + off; off += BB;

  gemm1_kernel<<<BB * (SS / 16), 256, 0, stream>>>(x, W1, a_src1, a_dst1,
                                                   h1, e_src1, e_dst1);
  attn1_kernel<<<BB * NH * 2, 128, 0, stream>>>(h1, e_src1, e_dst1, adj, b1, g1);
  gemm2_kernel<<<BB * SS / 16, 128, 0, stream>>>(g1, W2, h2);
  e2_kernel<<<(BB * SS + 255) / 256, 256, 0, stream>>>(h2, a_src2, a_dst2,
                                                       e_src2, e_dst_last);
  final_kernel<<<BB, 128, 0, stream>>>(h2, e_src2, e_dst_last, adj, b2, Wf, bfp, out);
}